// CustomModel_31361851195525
// MI455X (gfx1250) — compile-verified
//
#include <hip/hip_runtime.h>
#include <math.h>

typedef __attribute__((ext_vector_type(2))) float v2f;
typedef __attribute__((ext_vector_type(8))) float v8f;

#define SEQ 4096
#define DIM 256

// -ln(10000)/128
#define NEG_LOG1E4_OVER_HALF (-0.0719557893f)

// ---------------------------------------------------------------------------
// Kernel A: q[b][h] = sum_i x[b,0,i] * W0[i][h]   (RoPE at pos 0 == identity)
// ---------------------------------------------------------------------------
__global__ __launch_bounds__(256) void qproj_kernel(const float* __restrict__ x,
                                                    const float* __restrict__ W0,
                                                    float* __restrict__ q) {
    __shared__ float xrow[DIM];
    const int b = blockIdx.x;
    const int h = threadIdx.x;
    xrow[h] = x[(size_t)b * SEQ * DIM + h];
    __syncthreads();
    float acc = 0.f;
#pragma unroll 8
    for (int i = 0; i < DIM; ++i) acc += xrow[i] * W0[i * DIM + h];
    q[b * DIM + h] = acc;
}

// ---------------------------------------------------------------------------
// RoPE + dot-with-q on one 16x16 WMMA C tile, accumulating per-lane row sums.
// C layout (f32 16x16): lane L, vgpr r -> row (r + 8*(L>>4)), col (L&15).
// Pair partner for RoPE (adjacent column) is lane L^1 -> __shfl_xor(.,1).
// ---------------------------------------------------------------------------
__device__ __forceinline__ void rope_dot(v8f acc, int n0, int lane, int row0,
                                         const float* __restrict__ qb,
                                         float* __restrict__ sacc) {
    const int nn   = lane & 15;
    const int half = lane >> 4;
    const int h    = n0 + nn;
    const int p    = h >> 1;
    const bool odd = (h & 1) != 0;
    const float freq = __expf((float)p * NEG_LOG1E4_OVER_HALF);
    const float qh   = qb[h];
#pragma unroll
    for (int r = 0; r < 8; ++r) {
        float y    = acc[r];
        float mate = __shfl_xor(y, 1, 32);          // value of paired column
        const int t = row0 + r + 8 * half;          // global sequence position
        float s, c;
        __sincosf((float)t * freq, &s, &c);
        // even col: e*c - o*s   |  odd col: e*s + o*c
        float rot = odd ? (mate * s + y * c) : (y * c - mate * s);
        sacc[r] += rot * qh;
    }
}

// ---------------------------------------------------------------------------
// Kernel B: scores[b][t] = q_b . rope_t( X_b[t,:] @ W1 )
// Block = 128 threads (4 waves); each block computes 16 rows (one row tile),
// each wave covers 64 of the 256 output columns (4 col-tiles of 16).
// GEMM via v_wmma_f32_16x16x4_f32, K = 256 in 64 steps.
// ---------------------------------------------------------------------------
__global__ __launch_bounds__(128) void score_kernel(const float* __restrict__ x,
                                                    const float* __restrict__ W1,
                                                    const float* __restrict__ q,
                                                    float* __restrict__ scores) {
    const int b    = blockIdx.y;
    const int row0 = blockIdx.x * 16;
    const int lane = threadIdx.x & 31;
    const int wave = threadIdx.x >> 5;
    const int m    = lane & 15;
    const int half = lane >> 4;

    const float* __restrict__ Xb = x + (size_t)b * SEQ * DIM;
    // A(16x4) layout: lane L, vgpr v -> (row L&15, k = 2*(L>>4) + v): a contiguous float2
    const float* __restrict__ Arow = Xb + (size_t)(row0 + m) * DIM + 2 * half;
    // B(4x16) layout: lane L, vgpr v -> (k = 2*(L>>4) + v, col L&15)
    const int nbase = wave * 64 + m;  // column of this lane for ct=0 (+16 per ct)

    v8f acc0 = {}, acc1 = {}, acc2 = {}, acc3 = {};

    for (int k0 = 0; k0 < DIM; k0 += 4) {
        v2f a = *(const v2f*)(Arow + k0);
        const float* __restrict__ Wk = W1 + (size_t)(k0 + 2 * half) * DIM + nbase;
        v2f b0v = { Wk[0],       Wk[DIM]      };
        v2f b1v = { Wk[16],      Wk[16 + DIM] };
        v2f b2v = { Wk[32],      Wk[32 + DIM] };
        v2f b3v = { Wk[48],      Wk[48 + DIM] };
        acc0 = __builtin_amdgcn_wmma_f32_16x16x4_f32(false, a, false, b0v, (short)0, acc0, false, false);
        acc1 = __builtin_amdgcn_wmma_f32_16x16x4_f32(false, a, false, b1v, (short)0, acc1, false, false);
        acc2 = __builtin_amdgcn_wmma_f32_16x16x4_f32(false, a, false, b2v, (short)0, acc2, false, false);
        acc3 = __builtin_amdgcn_wmma_f32_16x16x4_f32(false, a, false, b3v, (short)0, acc3, false, false);
    }

    const float* __restrict__ qb = q + b * DIM;
    float sacc[8];
#pragma unroll
    for (int r = 0; r < 8; ++r) sacc[r] = 0.f;
    rope_dot(acc0, wave * 64 +  0, lane, row0, qb, sacc);
    rope_dot(acc1, wave * 64 + 16, lane, row0, qb, sacc);
    rope_dot(acc2, wave * 64 + 32, lane, row0, qb, sacc);
    rope_dot(acc3, wave * 64 + 48, lane, row0, qb, sacc);

    // Reduce over the 16 lanes of each half (cols) -> per-row partial sums
#pragma unroll
    for (int r = 0; r < 8; ++r) {
        float v = sacc[r];
        v += __shfl_xor(v, 1, 32);
        v += __shfl_xor(v, 2, 32);
        v += __shfl_xor(v, 4, 32);
        v += __shfl_xor(v, 8, 32);
        sacc[r] = v;
    }

    __shared__ float part[4][16];
    if ((lane & 15) == 0) {
#pragma unroll
        for (int r = 0; r < 8; ++r) part[wave][r + 8 * half] = sacc[r];
    }
    __syncthreads();
    if (threadIdx.x < 16) {
        float s = part[0][threadIdx.x] + part[1][threadIdx.x] +
                  part[2][threadIdx.x] + part[3][threadIdx.x];
        scores[b * SEQ + row0 + threadIdx.x] = s;
    }
}

// ---------------------------------------------------------------------------
// Kernel C: out[b,:] = ( softmax(scores[b,:]) @ X_b ) @ W2
// One block per batch, 256 threads (thread == hidden channel).
// ---------------------------------------------------------------------------
__global__ __launch_bounds__(256) void softmax_av_kernel(const float* __restrict__ x,
                                                         const float* __restrict__ W2,
                                                         const float* __restrict__ scores,
                                                         float* __restrict__ out) {
    __shared__ float attn[SEQ];     // 16 KB
    __shared__ float red[256];
    __shared__ float rbuf[DIM];
    const int b   = blockIdx.x;
    const int tid = threadIdx.x;

    // load scores + local max
    float lmax = -3.402823466e38f;
    for (int i = tid; i < SEQ; i += 256) {
        float v = scores[b * SEQ + i];
        attn[i] = v;
        lmax = fmaxf(lmax, v);
    }
    red[tid] = lmax;
    __syncthreads();
    for (int off = 128; off > 0; off >>= 1) {
        if (tid < off) red[tid] = fmaxf(red[tid], red[tid + off]);
        __syncthreads();
    }
    const float gmax = red[0];
    __syncthreads();

    // exp + sum
    float lsum = 0.f;
    for (int i = tid; i < SEQ; i += 256) {
        float e = __expf(attn[i] - gmax);
        attn[i] = e;
        lsum += e;
    }
    red[tid] = lsum;
    __syncthreads();
    for (int off = 128; off > 0; off >>= 1) {
        if (tid < off) red[tid] += red[tid + off];
        __syncthreads();
    }
    const float inv = 1.f / red[0];
    __syncthreads();

    // r[h] = sum_t attn[t] * X_b[t,h]  (coalesced over tid; LDS broadcast of attn)
    const float* __restrict__ Xb = x + (size_t)b * SEQ * DIM;
    float a0 = 0.f, a1 = 0.f, a2 = 0.f, a3 = 0.f;
    for (int t = 0; t < SEQ; t += 4) {
        a0 += attn[t + 0] * Xb[(size_t)(t + 0) * DIM + tid];
        a1 += attn[t + 1] * Xb[(size_t)(t + 1) * DIM + tid];
        a2 += attn[t + 2] * Xb[(size_t)(t + 2) * DIM + tid];
        a3 += attn[t + 3] * Xb[(size_t)(t + 3) * DIM + tid];
    }
    rbuf[tid] = ((a0 + a1) + (a2 + a3)) * inv;
    __syncthreads();

    // out[b][o] = sum_h rbuf[h] * W2[h][o]
    float o = 0.f;
#pragma unroll 8
    for (int h = 0; h < DIM; ++h) o += rbuf[h] * W2[h * DIM + tid];
    out[b * DIM + tid] = o;
}

// ---------------------------------------------------------------------------
extern "C" void kernel_launch(void* const* d_in, const int* in_sizes, int n_in,
                              void* d_out, int out_size, void* d_ws, size_t ws_size,
                              hipStream_t stream) {
    const float* x  = (const float*)d_in[0];   // (4, 4096, 256)
    const float* W0 = (const float*)d_in[1];   // (256, 256)
    const float* W1 = (const float*)d_in[2];   // (256, 256)
    const float* W2 = (const float*)d_in[3];   // (256, 256)
    float* out = (float*)d_out;                // (4, 256)

    float* qws = (float*)d_ws;                 // 4*256 floats
    float* sws = qws + 4 * DIM;                // 4*4096 floats

    qproj_kernel<<<4, 256, 0, stream>>>(x, W0, qws);
    score_kernel<<<dim3(SEQ / 16, 4), 128, 0, stream>>>(x, W1, qws, sws);
    softmax_av_kernel<<<4, 256, 0, stream>>>(x, W2, sws, out);
}